// GATEnc_54056458388022
// MI455X (gfx1250) — compile-verified
//
#include <hip/hip_runtime.h>
#include <hip/hip_bf16.h>
#include <cstddef>

typedef float v2f __attribute__((ext_vector_type(2)));
typedef float v8f __attribute__((ext_vector_type(8)));

#define NBATCH 16
#define FDIM   128
#define NEG_SLOPE 0.2f
#define AGG_TILE 128

// ---------------------------------------------------------------------------
// GEMM: C[M,Nc] = A[M,K] * B[K,Nc], fp32 row-major.
// One wave computes a 16x64 strip via V_WMMA_F32_16X16X4_F32 (A frag reused 4x).
// ---------------------------------------------------------------------------
__global__ void gat_gemm_wmma(const float* __restrict__ A,
                              const float* __restrict__ Bm,
                              float* __restrict__ C,
                              int M, int K, int Nc) {
    const int wavesPerBlock = blockDim.x >> 5;
    const int wave = blockIdx.x * wavesPerBlock + (threadIdx.x >> 5);
    const int lane = threadIdx.x & 31;
    const int ntiles = Nc >> 6;
    const int mt = wave / ntiles;
    const int ng = wave - mt * ntiles;
    if (mt * 16 >= M) return;                   // wave-uniform: EXEC stays all-1s

    const int m0 = mt << 4;
    const int n0 = ng << 6;
    const int rowA = m0 + (lane & 15);
    const int koff = (lane >> 4) << 1;          // 0 for lanes 0-15, 2 for 16-31
    const int colB = lane & 15;

    const float* Arow = A + (size_t)rowA * K + koff;

    v8f acc0 = {}, acc1 = {}, acc2 = {}, acc3 = {};

    for (int k0 = 0; k0 < K; k0 += 4) {
        v2f a;
        a.x = Arow[k0];
        a.y = Arow[k0 + 1];
        const float* Bbase = Bm + (size_t)(k0 + koff) * Nc + n0 + colB;
        v2f b0, b1, b2, b3;
        b0.x = Bbase[0];       b0.y = Bbase[Nc];
        b1.x = Bbase[16];      b1.y = Bbase[Nc + 16];
        b2.x = Bbase[32];      b2.y = Bbase[Nc + 32];
        b3.x = Bbase[48];      b3.y = Bbase[Nc + 48];
        acc0 = __builtin_amdgcn_wmma_f32_16x16x4_f32(false, a, false, b0, (short)0, acc0, false, false);
        acc1 = __builtin_amdgcn_wmma_f32_16x16x4_f32(false, a, false, b1, (short)0, acc1, false, false);
        acc2 = __builtin_amdgcn_wmma_f32_16x16x4_f32(false, a, false, b2, (short)0, acc2, false, false);
        acc3 = __builtin_amdgcn_wmma_f32_16x16x4_f32(false, a, false, b3, (short)0, acc3, false, false);
    }

    const int rbase = (lane < 16) ? 0 : 8;
    const int col = lane & 15;
    float* Cbase = C + (size_t)(m0 + rbase) * Nc + n0 + col;
#pragma unroll
    for (int r = 0; r < 8; ++r) {
        float* Crow = Cbase + (size_t)r * Nc;
        Crow[0]  = acc0[r];
        Crow[16] = acc1[r];
        Crow[32] = acc2[r];
        Crow[48] = acc3[r];
    }
}

// ---------------------------------------------------------------------------
// Per-node attention logits: alpha_s[n,h] = sum_f h[n,h,f]*att_src[h,f]
// ---------------------------------------------------------------------------
__global__ void gat_alpha(const float* __restrict__ h,
                          const float* __restrict__ asrc,
                          const float* __restrict__ adst,
                          float* __restrict__ alpha_s,
                          float* __restrict__ alpha_d,
                          int H) {
    __shared__ float s1[FDIM];
    __shared__ float s2[FDIM];
    const int n = blockIdx.x;
    const int t = threadIdx.x;
    for (int hd = 0; hd < H; ++hd) {
        float v = h[(size_t)n * H * FDIM + hd * FDIM + t];
        s1[t] = v * asrc[hd * FDIM + t];
        s2[t] = v * adst[hd * FDIM + t];
        __syncthreads();
        for (int s = 64; s > 0; s >>= 1) {
            if (t < s) { s1[t] += s1[t + s]; s2[t] += s2[t + s]; }
            __syncthreads();
        }
        if (t == 0) {
            alpha_s[n * H + hd] = s1[0];
            alpha_d[n * H + hd] = s2[0];
        }
        __syncthreads();
    }
}

// ---------------------------------------------------------------------------
// CSR build (once, shared by all 3 layers): deg histogram -> scan -> fill
// ---------------------------------------------------------------------------
__device__ __forceinline__ int edge_src(const int* ei, int E, int i) {
    return (i < E) ? ei[i] : (i - E);
}
__device__ __forceinline__ int edge_dst(const int* ei, int E, int i) {
    return (i < E) ? ei[E + i] : (i - E);
}

__global__ void gat_count(const int* __restrict__ ei, int E, int totE,
                          int* __restrict__ deg) {
    int i = blockIdx.x * blockDim.x + threadIdx.x;
    if (i >= totE) return;
    atomicAdd(&deg[edge_dst(ei, E, i)], 1);
}

// single-block exclusive scan (n up to ~hundreds of K is fine)
__global__ void gat_scan(const int* __restrict__ deg, int* __restrict__ off, int n) {
    __shared__ int buf[1024];
    __shared__ int carry;
    const int t = threadIdx.x;
    if (t == 0) carry = 0;
    __syncthreads();
    for (int base = 0; base < n; base += 1024) {
        int i = base + t;
        int v = (i < n) ? deg[i] : 0;
        buf[t] = v;
        __syncthreads();
        for (int s = 1; s < 1024; s <<= 1) {
            int add = (t >= s) ? buf[t - s] : 0;
            __syncthreads();
            buf[t] += add;
            __syncthreads();
        }
        if (i < n) off[i] = carry + buf[t] - v;   // exclusive
        int total = buf[1023];
        __syncthreads();
        if (t == 0) carry += total;
        __syncthreads();
    }
    if (t == 0) off[n] = carry;
}

__global__ void gat_fill(const int* __restrict__ ei, int E, int totE,
                         int* __restrict__ cursor, int* __restrict__ csr) {
    int i = blockIdx.x * blockDim.x + threadIdx.x;
    if (i >= totE) return;
    int d = edge_dst(ei, E, i);
    int pos = atomicAdd(&cursor[d], 1);
    csr[pos] = i;
}

// ---------------------------------------------------------------------------
// Segment softmax over incoming edges, atomic-free: one thread per (dst, head).
// Writes normalized weight per edge into wbuf[i*H+hd].
// ---------------------------------------------------------------------------
__global__ void gat_softmax_csr(const int* __restrict__ off,
                                const int* __restrict__ csr,
                                const int* __restrict__ ei, int E,
                                const float* __restrict__ as,
                                const float* __restrict__ ad,
                                float* __restrict__ wbuf,
                                int N, int H) {
    int idx = blockIdx.x * blockDim.x + threadIdx.x;
    if (idx >= N * H) return;
    int d = idx / H, hd = idx - d * H;
    const float adv = ad[d * H + hd];
    const int kb = off[d], ke = off[d + 1];
    float m = -1e30f;
    for (int k = kb; k < ke; ++k) {
        int i = csr[k];
        float e = as[edge_src(ei, E, i) * H + hd] + adv;
        e = (e > 0.f) ? e : NEG_SLOPE * e;
        m = fmaxf(m, e);
    }
    float sum = 0.f;
    for (int k = kb; k < ke; ++k) {
        int i = csr[k];
        float e = as[edge_src(ei, E, i) * H + hd] + adv;
        e = (e > 0.f) ? e : NEG_SLOPE * e;
        float ex = expf(e - m);
        wbuf[i * H + hd] = ex;
        sum += ex;
    }
    float inv = 1.f / sum;
    for (int k = kb; k < ke; ++k) {
        wbuf[csr[k] * H + hd] *= inv;
    }
}

// ---------------------------------------------------------------------------
// Weighted aggregation, atomic-free, with fused bias (+ReLU):
// one 128-thread block per dst; (src, w[H]) tiles staged in LDS;
// each thread accumulates HF/128 features in registers, single store.
// ---------------------------------------------------------------------------
__global__ void gat_agg_csr(const int* __restrict__ off,
                            const int* __restrict__ csr,
                            const int* __restrict__ ei, int E,
                            const float* __restrict__ wbuf,
                            const float* __restrict__ h,
                            const float* __restrict__ bias,
                            float* __restrict__ out,
                            int H, int HF, int doRelu) {
    __shared__ int   s_src[AGG_TILE];
    __shared__ float s_w[AGG_TILE * 4];         // H <= 4
    const int d = blockIdx.x;
    const int t = threadIdx.x;                  // 0..127
    const int vec = HF >> 7;                    // 4 (HF=512) or 1 (HF=128)
    const int f0 = t * vec;
    const int hd = f0 / FDIM;
    float acc0 = 0.f, acc1 = 0.f, acc2 = 0.f, acc3 = 0.f;
    const int kb = off[d], ke = off[d + 1];
    for (int k0 = kb; k0 < ke; k0 += AGG_TILE) {
        int cnt = ke - k0; if (cnt > AGG_TILE) cnt = AGG_TILE;
        if (t < cnt) {
            int i = csr[k0 + t];
            s_src[t] = edge_src(ei, E, i);
            for (int hh = 0; hh < H; ++hh) s_w[t * H + hh] = wbuf[i * H + hh];
        }
        __syncthreads();
        if (vec == 4) {
            for (int k = 0; k < cnt; ++k) {
                float w = s_w[k * H + hd];
                const float4 hv = *(const float4*)(h + (size_t)s_src[k] * HF + f0);
                acc0 += hv.x * w; acc1 += hv.y * w; acc2 += hv.z * w; acc3 += hv.w * w;
            }
        } else {
            for (int k = 0; k < cnt; ++k) {
                acc0 += h[(size_t)s_src[k] * HF + f0] * s_w[k * H + hd];
            }
        }
        __syncthreads();
    }
    float* op = out + (size_t)d * HF + f0;
    float r0 = acc0 + bias[f0];
    if (doRelu) r0 = fmaxf(r0, 0.f);
    op[0] = r0;
    if (vec == 4) {
        float r1 = acc1 + bias[f0 + 1];
        float r2 = acc2 + bias[f0 + 2];
        float r3 = acc3 + bias[f0 + 3];
        if (doRelu) { r1 = fmaxf(r1, 0.f); r2 = fmaxf(r2, 0.f); r3 = fmaxf(r3, 0.f); }
        op[1] = r1; op[2] = r2; op[3] = r3;
    }
}

// ---------------------------------------------------------------------------
__global__ void gat_any_mask(const int* __restrict__ mask, int n,
                             int* __restrict__ flag) {
    int i = blockIdx.x * blockDim.x + threadIdx.x;
    if (i < n && mask[i]) atomicOr(flag, 1);
}

// masked global max pool: one block per batch, one thread per feature
__global__ void gat_pool(const float* __restrict__ h3,
                         const int* __restrict__ batch,
                         const int* __restrict__ mask,
                         const int* __restrict__ flag,
                         float* __restrict__ out, int n) {
    const int b = blockIdx.x;
    const int f = threadIdx.x;
    const int hasM = *flag;
    float best = -1e30f;
    for (int i = 0; i < n; ++i) {
        if (batch[i] != b) continue;
        if (mask[i] && hasM) continue;
        best = fmaxf(best, h3[(size_t)i * FDIM + f]);
    }
    out[b * FDIM + f] = best;
}

// ---------------------------------------------------------------------------
static void run_gat_layer(const float* hin, int K,
                          const float* W, const float* asrc, const float* adst,
                          const float* bias, int H,
                          const int* ei, int E, int N,
                          const int* off, const int* csr,
                          float* hmid, float* hout,
                          float* alpha_s, float* alpha_d, float* wbuf,
                          int doRelu, hipStream_t stream) {
    const int HF = H * FDIM;
    {   // 1) GEMM (WMMA)
        int totalWaves = (N / 16) * (HF / 64);
        int threads = totalWaves * 32;
        gat_gemm_wmma<<<(threads + 255) / 256, 256, 0, stream>>>(hin, W, hmid, N, K, HF);
    }
    // 2) attention logits
    gat_alpha<<<N, FDIM, 0, stream>>>(hmid, asrc, adst, alpha_s, alpha_d, H);
    // 3) segment softmax (atomic-free)
    {
        int work = N * H;
        gat_softmax_csr<<<(work + 255) / 256, 256, 0, stream>>>(
            off, csr, ei, E, alpha_s, alpha_d, wbuf, N, H);
    }
    // 4) weighted aggregation + bias (+ReLU), atomic-free
    gat_agg_csr<<<N, 128, 0, stream>>>(off, csr, ei, E, wbuf, hmid, bias, hout,
                                       H, HF, doRelu);
}

extern "C" void kernel_launch(void* const* d_in, const int* in_sizes, int n_in,
                              void* d_out, int out_size, void* d_ws, size_t ws_size,
                              hipStream_t stream) {
    const float* x        = (const float*)d_in[0];      // [N,128]
    const int*   ei       = (const int*)d_in[1];        // [2,E]
    const int*   batch_id = (const int*)d_in[2];        // [N]
    const int*   nmask    = (const int*)d_in[3];        // [N] bool
    // d_in[4] = edge_mask (ignored: edge_dim=None)
    const float* W1  = (const float*)d_in[5];
    const float* as1 = (const float*)d_in[6];
    const float* ad1 = (const float*)d_in[7];
    const float* b1  = (const float*)d_in[8];
    const float* W2  = (const float*)d_in[9];
    const float* as2 = (const float*)d_in[10];
    const float* ad2 = (const float*)d_in[11];
    const float* b2  = (const float*)d_in[12];
    const float* W3  = (const float*)d_in[13];
    const float* as3 = (const float*)d_in[14];
    const float* ad3 = (const float*)d_in[15];
    const float* b3  = (const float*)d_in[16];

    const int N = in_sizes[0] / FDIM;
    const int E = in_sizes[1] / 2;
    const int totE = E + N;                    // with self-loops

    // workspace carve-out
    float* hA      = (float*)d_ws;                         // N*512
    float* hB      = hA + (size_t)N * 512;                 // N*512
    float* alpha_s = hB + (size_t)N * 512;                 // N*4
    float* alpha_d = alpha_s + (size_t)N * 4;              // N*4
    float* wbuf    = alpha_d + (size_t)N * 4;              // totE*4
    int*   deg     = (int*)(wbuf + (size_t)totE * 4);      // N
    int*   off     = deg + N;                              // N+1
    int*   cursor  = off + (N + 1);                        // N
    int*   csr     = cursor + N;                           // totE
    int*   flag    = csr + totE;                           // 1

    // ---- CSR build (reused by all 3 layers) ----
    hipMemsetAsync(deg, 0, (size_t)N * sizeof(int), stream);
    gat_count<<<(totE + 255) / 256, 256, 0, stream>>>(ei, E, totE, deg);
    gat_scan<<<1, 1024, 0, stream>>>(deg, off, N);
    hipMemcpyAsync(cursor, off, (size_t)N * sizeof(int),
                   hipMemcpyDeviceToDevice, stream);
    gat_fill<<<(totE + 255) / 256, 256, 0, stream>>>(ei, E, totE, cursor, csr);

    // ---- layers ----
    run_gat_layer(x, FDIM, W1, as1, ad1, b1, 4, ei, E, N, off, csr,
                  hA, hB, alpha_s, alpha_d, wbuf, 1, stream);
    run_gat_layer(hB, 512, W2, as2, ad2, b2, 4, ei, E, N, off, csr,
                  hA, hB, alpha_s, alpha_d, wbuf, 1, stream);
    run_gat_layer(hB, 512, W3, as3, ad3, b3, 1, ei, E, N, off, csr,
                  hA, hB, alpha_s, alpha_d, wbuf, 0, stream);

    // ---- masked global max pool ----
    hipMemsetAsync(flag, 0, sizeof(int), stream);
    gat_any_mask<<<(N + 255) / 256, 256, 0, stream>>>(nmask, N, flag);
    gat_pool<<<NBATCH, FDIM, 0, stream>>>(hB, batch_id, nmask, flag, (float*)d_out, N);
}